// BayesLinear_42167988912911
// MI455X (gfx1250) — compile-verified
//
#include <hip/hip_runtime.h>
#include <cstddef>
#include <cstdint>

// ---------------------------------------------------------------------------
// BayesLinear: out[n,o] = sum_i x[n,i]*(mu_w[i,o] + exp(lw[i,o])*eps_w[n,i,o])
//                         + mu_b[o] + exp(lb[o])*eps_b[n,o]
// N=4096, IN=256, OUT=256.
// Bound: eps_w = 1 GiB streamed once -> ~46us at 23.3 TB/s. Compute is noise.
// Kernel plan:
//   K0: sig_w = exp(lw)                       (256 KB into d_ws)
//   K1: out = x @ mu_w + bias terms           (true GEMM -> fp32 WMMA 16x16x4)
//   K2: out += sum_i (x[n,i]*sig_w[i,o]) * eps_w[n,i,o]   (NT streaming)
// ---------------------------------------------------------------------------

typedef __attribute__((ext_vector_type(2))) float v2f;
typedef __attribute__((ext_vector_type(4))) float v4f;
typedef __attribute__((ext_vector_type(8))) float v8f;

#define NSAMP   4096
#define IN_DIM  256
#define OUT_DIM 256

// ---------------------------------------------------------------------------
// K0: sigma_w = exp(log_sigma_w), 65536 elements -> d_ws
// ---------------------------------------------------------------------------
__global__ void __launch_bounds__(256)
bayes_sigw_kernel(const float* __restrict__ logw, float* __restrict__ sig) {
    const int idx = blockIdx.x * 256 + threadIdx.x;
    sig[idx] = expf(logw[idx]);
}

// ---------------------------------------------------------------------------
// K1: out[n,o] = sum_k x[n,k]*mu[k,o] + b_mu[o] + exp(logb[o])*eps_b[n,o]
// One 16x16 C tile per wave via V_WMMA_F32_16X16X4_F32 (fp32-exact).
// Block = 8 waves = 128x16 output tile; grid = (4096/128)*16 = 512 blocks.
//
// ISA VGPR layouts (CDNA5 ISA):
//   A 16x4 f32 : lane L holds row M=L%16; VGPR0/1 = K = 2*(L>>4) + {0,1}
//   B 4x16 f32 : lane L holds col N=L%16; VGPR0/1 = K = 2*(L>>4) + {0,1}
//   C/D        : VGPR j holds M = j + 8*(L>>4), N = L%16
// ---------------------------------------------------------------------------
__global__ void __launch_bounds__(256)
bayes_mu_gemm_bias_kernel(const float* __restrict__ x,
                          const float* __restrict__ mu,
                          const float* __restrict__ bmu,
                          const float* __restrict__ logb,
                          const float* __restrict__ eps_b,
                          float* __restrict__ out) {
    const int tid  = threadIdx.x;
    const int lane = tid & 31;
    const int wave = tid >> 5;
    const int half = lane >> 4;    // 0 or 1
    const int lm   = lane & 15;

    const int m0  = ((blockIdx.x >> 4) << 7) + (wave << 4);  // row base of tile
    const int n0  = (blockIdx.x & 15) << 4;                  // col base of tile
    const int row = m0 + lm;                                 // A row for this lane
    const int n   = n0 + lm;                                 // B/C col for this lane

    v8f c = {};

#if __has_builtin(__builtin_amdgcn_wmma_f32_16x16x4_f32)
    #pragma unroll 8
    for (int kb = 0; kb < IN_DIM; kb += 4) {
        const int k0 = kb + 2 * half;
        v2f a = *(const v2f*)(x + row * IN_DIM + k0);   // contiguous K pair
        v2f b;
        b.x = mu[k0 * OUT_DIM + n];
        b.y = mu[(k0 + 1) * OUT_DIM + n];
        // 8 args: (neg_a, A, neg_b, B, c_mod, C, reuse_a, reuse_b)
        c = __builtin_amdgcn_wmma_f32_16x16x4_f32(false, a, false, b,
                                                  (short)0, c, false, false);
    }
#else
    // Scalar fallback (keeps compile green if builtin name differs; histogram
    // wmma==0 will flag it).
    #pragma unroll
    for (int j = 0; j < 8; ++j) {
        const int m = m0 + j + 8 * half;
        float acc = 0.0f;
        for (int k = 0; k < IN_DIM; ++k)
            acc = fmaf(x[m * IN_DIM + k], mu[k * OUT_DIM + n], acc);
        c[j] = acc;
    }
#endif

    const float sb = expf(logb[n]);   // one transcendental per lane
    const float bb = bmu[n];
    #pragma unroll
    for (int j = 0; j < 8; ++j) {
        const int m = m0 + j + 8 * half;
        out[m * OUT_DIM + n] = c[j] + bb + sb * eps_b[m * OUT_DIM + n];
    }
}

// ---------------------------------------------------------------------------
// K2: out[n,o] += sum_i (x[n,i]*sig[i,o]) * eps_w[n,i,o]
// 4 samples per 256-thread block; 64 threads per sample, float4 per thread.
// eps_w via non-temporal loads (1 GiB one-shot stream, keep L2 for sig/mu).
// ---------------------------------------------------------------------------
__global__ void __launch_bounds__(256)
bayes_eps_stream_kernel(const float* __restrict__ x,
                        const float* __restrict__ eps_w,
                        const float* __restrict__ sig,
                        float* __restrict__ out) {
    __shared__ float xs[4 * IN_DIM];   // 4 KB: one x row per sample
    const int tid = threadIdx.x;
    const int s   = tid >> 6;          // sample slot 0..3
    const int to  = tid & 63;          // output-quad index 0..63
    const int n   = blockIdx.x * 4 + s;
    const int o4  = to * 4;

    // Stage x[n,:] into LDS (64 threads x float4 = 256 floats per sample).
    *(v4f*)(&xs[s * IN_DIM + o4]) = *(const v4f*)(x + (size_t)n * IN_DIM + o4);
    __syncthreads();

    const float* ep = eps_w + (size_t)n * (IN_DIM * OUT_DIM) + o4;
    const float* sg = sig + o4;

    v4f acc = {};
    #pragma unroll 4
    for (int i = 0; i < IN_DIM; ++i) {
        const float xi = xs[s * IN_DIM + i];                     // LDS broadcast
        v4f e = __builtin_nontemporal_load((const v4f*)(ep + (size_t)i * OUT_DIM));
        v4f sv = *(const v4f*)(sg + i * OUT_DIM);                // L2/WGP$-hot
        acc += (xi * sv) * e;
    }

    v4f* po = (v4f*)(out + (size_t)n * OUT_DIM + o4);
    *po += acc;
}

// ---------------------------------------------------------------------------
// Launch
// ---------------------------------------------------------------------------
extern "C" void kernel_launch(void* const* d_in, const int* in_sizes, int n_in,
                              void* d_out, int out_size, void* d_ws, size_t ws_size,
                              hipStream_t stream) {
    const float* x     = (const float*)d_in[0];  // [4096,256]
    const float* eps_w = (const float*)d_in[1];  // [4096,256,256]
    const float* eps_b = (const float*)d_in[2];  // [4096,256]
    const float* w_mu  = (const float*)d_in[3];  // [256,256]
    const float* logw  = (const float*)d_in[4];  // [256,256]
    const float* b_mu  = (const float*)d_in[5];  // [256]
    const float* logb  = (const float*)d_in[6];  // [256]
    float* out = (float*)d_out;                  // [4096,256]
    float* sig = (float*)d_ws;                   // needs 256 KB scratch

    (void)in_sizes; (void)n_in; (void)out_size; (void)ws_size;

    bayes_sigw_kernel<<<(IN_DIM * OUT_DIM) / 256, 256, 0, stream>>>(logw, sig);
    bayes_mu_gemm_bias_kernel<<<512, 256, 0, stream>>>(x, w_mu, b_mu, logb,
                                                       eps_b, out);
    bayes_eps_stream_kernel<<<NSAMP / 4, 256, 0, stream>>>(x, eps_w, sig, out);
}